// SparseAttention_22385369547084
// MI455X (gfx1250) — compile-verified
//
#include <hip/hip_runtime.h>
#include <hip/hip_bf16.h>

// Sizes fixed by the reference
#define BB 2
#define SS 4096
#define DD 1024
#define HH 16
#define DH 64
#define WW 128
#define NW 32          // SS / WW
#define ROWS (BB*SS)   // 8192

typedef __attribute__((ext_vector_type(2))) float v2f;
typedef __attribute__((ext_vector_type(8))) float v8f;
typedef int v4i __attribute__((vector_size(4 * sizeof(int))));

static __device__ __forceinline__ v8f wmma4(v2f a, v2f b, v8f c) {
    // V_WMMA_F32_16X16X4_F32 : D = A(16x4) * B(4x16) + C(16x16), fp32 exact
    return __builtin_amdgcn_wmma_f32_16x16x4_f32(
        false, a, false, b, (short)0, c, false, false);
}

// ---- CDNA5 async global->LDS staging (ASYNCcnt path) ----------------------
#if defined(__has_builtin)
#if __has_builtin(__builtin_amdgcn_global_load_async_to_lds_b128)
#define ASYNC_COPY 1
#endif
#endif

static __device__ __forceinline__ void async_wait_le4() {
#ifdef ASYNC_COPY
#if __has_builtin(__builtin_amdgcn_s_wait_asynccnt)
    __builtin_amdgcn_s_wait_asynccnt(4);
#else
    asm volatile("s_wait_asynccnt 0x4" ::: "memory");
#endif
#endif
}
static __device__ __forceinline__ void async_wait_le0() {
#ifdef ASYNC_COPY
#if __has_builtin(__builtin_amdgcn_s_wait_asynccnt)
    __builtin_amdgcn_s_wait_asynccnt(0);
#else
    asm volatile("s_wait_asynccnt 0x0" ::: "memory");
#endif
#endif
}

// Stage a 64(k) x 64(n) fp32 panel of B into LDS (row stride 80 floats).
// 1024 x 16B transfers, 256 threads -> 4 per thread (4 async instr per wave).
static __device__ __forceinline__ void stage_b(
    const float* __restrict__ Bg, int N, int n0, int k0, float* Bs) {
    int t = threadIdx.x;
#pragma unroll
    for (int i = 0; i < 4; ++i) {
        int id = i * 256 + t;
        int row = id >> 4;
        int coll = (id & 15) * 4;
        const float* src = Bg + (size_t)(k0 + row) * N + n0 + coll;
        float* dst = Bs + row * 80 + coll;
#ifdef ASYNC_COPY
        __builtin_amdgcn_global_load_async_to_lds_b128(
            (v4i*)src, (v4i*)dst, 0, 0);
#else
        *(float4*)dst = *(const float4*)src;
#endif
    }
}

// ---------------------------------------------------------------------------
// Kernel 1: RMSNorm  x = inp * rsqrt(mean(inp^2) + eps) * rms_w
// ---------------------------------------------------------------------------
__global__ __launch_bounds__(256) void rmsnorm_kernel(
    const float* __restrict__ inp, const float* __restrict__ w,
    float* __restrict__ x) {
    int row = blockIdx.x;
    const float* in = inp + (size_t)row * DD;
    float ss = 0.f;
    for (int i = threadIdx.x; i < DD; i += 256) { float v = in[i]; ss += v * v; }
    for (int off = 16; off > 0; off >>= 1) ss += __shfl_xor(ss, off, 32);
    __shared__ float red[8];
    int wid = threadIdx.x >> 5;
    if ((threadIdx.x & 31) == 0) red[wid] = ss;
    __syncthreads();
    float tot = red[0] + red[1] + red[2] + red[3] + red[4] + red[5] + red[6] + red[7];
    float rs = rsqrtf(tot * (1.0f / (float)DD) + 1.1920929e-07f);
    float* out = x + (size_t)row * DD;
    for (int i = threadIdx.x; i < DD; i += 256) out[i] = in[i] * rs * w[i];
}

// ---------------------------------------------------------------------------
// Kernel 2: gates  g[row][h] = sum_j sigmoid((x @ strat_w)[row][h*3+j] + b)
// ---------------------------------------------------------------------------
__global__ __launch_bounds__(64) void gates_kernel(
    const float* __restrict__ x, const float* __restrict__ sw,
    const float* __restrict__ sb, float* __restrict__ g) {
    int row = blockIdx.x;
    int c = threadIdx.x;
    __shared__ float zs[48];
    if (c < 48) {
        const float* xr = x + (size_t)row * DD;
        float acc = sb[c];
        for (int d = 0; d < DD; ++d) acc = fmaf(xr[d], sw[(size_t)d * 48 + c], acc);
        zs[c] = 1.f / (1.f + __expf(-acc));
    }
    __syncthreads();
    if (c < HH) g[(size_t)row * HH + c] = zs[c * 3] + zs[c * 3 + 1] + zs[c * 3 + 2];
}

// ---------------------------------------------------------------------------
// Kernel 3: fp32 WMMA GEMM  C[M,N] = A[M,K] @ B[K,N]  (row-major)
// 256 threads = 8 waves; macro tile 256x64; wave tile 32x64 (8 accumulators).
// B panel (64 x 64) double-buffered in LDS via async-to-LDS, stride 80 floats
// so the two lane-halves (rows k, k+2) hit disjoint bank sets.
// ---------------------------------------------------------------------------
__global__ __launch_bounds__(256) void gemm_wmma_f32(
    const float* __restrict__ A, const float* __restrict__ Bm,
    float* __restrict__ C, int M, int N, int K) {
    __shared__ __align__(16) float Bs[2][64][80];

    int tid = threadIdx.x;
    int lane = tid & 31;
    int wave = tid >> 5;
    int nl = lane & 15;
    int hl = lane >> 4;
    int kb2 = hl * 2;
    int m0 = blockIdx.y * 256;
    int n0 = blockIdx.x * 64;
    int mw = m0 + wave * 32;

    v8f z = {};
    v8f acc[2][4];
#pragma unroll
    for (int u = 0; u < 2; ++u)
#pragma unroll
        for (int j = 0; j < 4; ++j) acc[u][j] = z;

    const float* arow0 = A + (size_t)(mw + nl) * K + kb2;
    const float* arow1 = arow0 + (size_t)16 * K;

    const int NC = K / 64;
    stage_b(Bm, N, n0, 0, &Bs[0][0][0]);

    for (int c = 0; c < NC; ++c) {
        if (c + 1 < NC) {
            stage_b(Bm, N, n0, (c + 1) * 64, &Bs[(c + 1) & 1][0][0]);
            async_wait_le4();   // previous chunk's 4 async ops complete (in-order)
        } else {
            async_wait_le0();
        }
        __syncthreads();

        const float(*Bb)[80] = Bs[c & 1];
        int k0 = c * 64;
#pragma unroll 4
        for (int s = 0; s < 16; ++s) {
            int kk = s * 4;
            v2f a0; a0.x = arow0[k0 + kk]; a0.y = arow0[k0 + kk + 1];
            v2f a1; a1.x = arow1[k0 + kk]; a1.y = arow1[k0 + kk + 1];
#pragma unroll
            for (int j = 0; j < 4; ++j) {
                v2f b;
                b.x = Bb[kk + kb2][j * 16 + nl];
                b.y = Bb[kk + kb2 + 1][j * 16 + nl];
                acc[0][j] = wmma4(a0, b, acc[0][j]);
                acc[1][j] = wmma4(a1, b, acc[1][j]);
            }
        }
        __syncthreads();   // all waves done reading before buffer is re-filled
    }

#pragma unroll
    for (int u = 0; u < 2; ++u)
#pragma unroll
        for (int j = 0; j < 4; ++j)
#pragma unroll
            for (int r = 0; r < 8; ++r)
                C[(size_t)(mw + u * 16 + r + 8 * hl) * N + n0 + j * 16 + nl] =
                    acc[u][j][r];
}

// ---------------------------------------------------------------------------
// Kernel 4: sliding-window causal attention + gate multiply.
// qkv layout: [row(b,s)][3072]; q at h*64+d, k at 1024+..., v at 2048+...
// Block = 64 threads (2 waves); wave = 16 queries vs 256 keys.
// ---------------------------------------------------------------------------
__global__ __launch_bounds__(64) void attn_kernel(
    const float* __restrict__ qkv, const float* __restrict__ g,
    float* __restrict__ outp) {
    int lane = threadIdx.x & 31;
    int wave = threadIdx.x >> 5;
    int nl = lane & 15;
    int hl = lane >> 4;
    int kb2 = hl * 2;

    int sub = blockIdx.x;          // 0..3
    int n0  = blockIdx.y;          // 0..31 window block
    int bh  = blockIdx.z;          // 0..31
    int b = bh >> 4, h = bh & 15;
    int wqBase = sub * 32 + wave * 16;
    int kbase = n0 * WW - WW;

    // Q fragments: A(16x64) -> 16 chunks of K=4
    int srow_a = n0 * WW + wqBase + nl;
    const float* qr = qkv + ((size_t)(b * SS + srow_a)) * 3072 + h * 64;
    v2f afr[16];
#pragma unroll
    for (int t = 0; t < 16; ++t) { afr[t].x = qr[t * 4 + kb2]; afr[t].y = qr[t * 4 + kb2 + 1]; }

    // Scores: 16 C-tiles (16 queries x 256 keys)
    v8f sarr[16];
    v8f z = {};
#pragma unroll
    for (int t = 0; t < 16; ++t) {
        v8f sc = z;
        int nn = t * 16 + nl;
        int kpos = kbase + nn;
        int kclamp = kpos < 0 ? 0 : kpos;
        const float* kp = qkv + ((size_t)(b * SS + kclamp)) * 3072 + 1024 + h * 64;
        float vsel = (kpos >= 0) ? 1.f : 0.f;
#pragma unroll
        for (int d0 = 0; d0 < 16; ++d0) {
            v2f bf;
            bf.x = kp[d0 * 4 + kb2] * vsel;
            bf.y = kp[d0 * 4 + kb2 + 1] * vsel;
            sc = wmma4(afr[d0], bf, sc);
        }
        sarr[t] = sc;
    }

    // Masked softmax; rows live in 16-lane halves of the wave.
    __shared__ float attn_s[2][16][256];
#pragma unroll
    for (int r = 0; r < 8; ++r) {
        int mq = wqBase + r + 8 * hl;
        float mx = -3.0e38f;
#pragma unroll
        for (int t = 0; t < 16; ++t) {
            int nn = t * 16 + nl;
            int kpos = kbase + nn;
            bool ok = (kpos >= 0) && (nn >= mq) && (nn <= mq + WW);
            float v = ok ? sarr[t][r] * 0.125f : -3.0e38f;
            sarr[t][r] = v;
            mx = fmaxf(mx, v);
        }
        mx = fmaxf(mx, __shfl_xor(mx, 1, 16));
        mx = fmaxf(mx, __shfl_xor(mx, 2, 16));
        mx = fmaxf(mx, __shfl_xor(mx, 4, 16));
        mx = fmaxf(mx, __shfl_xor(mx, 8, 16));
        float sum = 0.f;
#pragma unroll
        for (int t = 0; t < 16; ++t) {
            float p = __expf(sarr[t][r] - mx);
            sarr[t][r] = p;
            sum += p;
        }
        sum += __shfl_xor(sum, 1, 16);
        sum += __shfl_xor(sum, 2, 16);
        sum += __shfl_xor(sum, 4, 16);
        sum += __shfl_xor(sum, 8, 16);
        float inv = 1.f / sum;
#pragma unroll
        for (int t = 0; t < 16; ++t)
            attn_s[wave][r + 8 * hl][t * 16 + nl] = sarr[t][r] * inv;
    }
    __syncthreads();

    // out(16x64) = attn(16x256) @ V(256x64)
    v8f oacc[4];
    oacc[0] = z; oacc[1] = z; oacc[2] = z; oacc[3] = z;
    const float* vbase = qkv + (size_t)(b * SS) * 3072 + 2048 + h * 64;
    for (int kc = 0; kc < 64; ++kc) {
        int kk = kc * 4 + kb2;
        v2f a;
        a.x = attn_s[wave][nl][kk];
        a.y = attn_s[wave][nl][kk + 1];
        int kpx = kbase + kk;
        int kpy = kpx + 1;
        const float* vx = vbase + (size_t)(kpx < 0 ? 0 : kpx) * 3072;
        const float* vy = vbase + (size_t)(kpy < 0 ? 0 : kpy) * 3072;
        float sx = (kpx >= 0) ? 1.f : 0.f;
        float sy = (kpy >= 0) ? 1.f : 0.f;
#pragma unroll
        for (int t = 0; t < 4; ++t) {
            v2f bf;
            bf.x = vx[t * 16 + nl] * sx;
            bf.y = vy[t * 16 + nl] * sy;
            oacc[t] = wmma4(a, bf, oacc[t]);
        }
    }

#pragma unroll
    for (int r = 0; r < 8; ++r) {
        int m = r + 8 * hl;
        int srow = n0 * WW + wqBase + m;
        float gate = g[(size_t)(b * SS + srow) * HH + h];
#pragma unroll
        for (int t = 0; t < 4; ++t)
            outp[(size_t)(b * SS + srow) * DD + h * 64 + t * 16 + nl] = oacc[t][r] * gate;
    }
}

// ---------------------------------------------------------------------------
extern "C" void kernel_launch(void* const* d_in, const int* in_sizes, int n_in,
                              void* d_out, int out_size, void* d_ws, size_t ws_size,
                              hipStream_t stream) {
    const float* inp     = (const float*)d_in[0];
    const float* rms_w   = (const float*)d_in[1];
    const float* wqkv    = (const float*)d_in[2];
    // d_in[3..8] dead branch (k_pos, v_pos, kc_*, vc_*)
    const float* strat_w = (const float*)d_in[9];
    const float* strat_b = (const float*)d_in[10];
    const float* out_w   = (const float*)d_in[11];
    float* out = (float*)d_out;

    float* ws   = (float*)d_ws;
    float* x    = ws;                                  // 8192*1024
    float* qkv  = x + (size_t)ROWS * DD;               // 8192*3072
    float* gbuf = qkv + (size_t)ROWS * 3 * DD;         // 8192*16
    float* pre  = gbuf + (size_t)ROWS * HH;            // 8192*1024

    rmsnorm_kernel<<<ROWS, 256, 0, stream>>>(inp, rms_w, x);
    gates_kernel<<<ROWS, 64, 0, stream>>>(x, strat_w, strat_b, gbuf);
    gemm_wmma_f32<<<dim3(3 * DD / 64, ROWS / 256), 256, 0, stream>>>(
        x, wqkv, qkv, ROWS, 3 * DD, DD);
    attn_kernel<<<dim3(WW / 32, NW, BB * HH), 64, 0, stream>>>(qkv, gbuf, pre);
    gemm_wmma_f32<<<dim3(DD / 64, ROWS / 256), 256, 0, stream>>>(
        pre, out_w, out, ROWS, DD, DD);
}